// LieGroupProcessor_30580167147874
// MI455X (gfx1250) — compile-verified
//
#include <hip/hip_runtime.h>

typedef __attribute__((ext_vector_type(2))) float v2f;
typedef __attribute__((ext_vector_type(8))) float v8f;

#define SUBSP      10
#define ZCOLS      100          // 10 subgroups * 10 subspace
#define BATCH      65536
#define TAYLOR_N   8
#define SQUARINGS  5
#define INV_SCALE  (1.0f / 32.0f)   // 2^-SQUARINGS
#define NBLOCKS    2560             // 20480 waves; MUST keep waves % 5 == 0

__device__ __forceinline__ void wave_lds_fence() {
    // LDS is in-order within a wave; this stops compiler reordering and
    // drains DScnt before cross-lane reads of freshly stored tile data.
    asm volatile("s_wait_dscnt 0" ::: "memory");
}

__global__ __launch_bounds__(256) void lie_expm_kernel(
    const float* __restrict__ z,
    const float* __restrict__ basis,
    float* __restrict__ out)
{
    __shared__ float sT[8][256];      // per-wave 16x16 f32 scratch tile

    const int lane = threadIdx.x & 31;
    const int wave = threadIdx.x >> 5;
    float* S = sT[wave];

    const int hlf = lane >> 4;        // which half of the wave
    const int lx  = lane & 15;

    // Identity matrix in WMMA D-layout: d[r] is element (m=r+8*hlf, n=lx)
    float idv[8];
    #pragma unroll
    for (int r = 0; r < 8; ++r)
        idv[r] = ((r + 8 * hlf) == lx) ? 1.0f : 0.0f;

    const int wavesTotal = gridDim.x * 8;     // multiple of 5 by construction
    const int gw = blockIdx.x * 8 + wave;

    // Subgroup-pair index is invariant for this wave (wavesTotal % 5 == 0).
    const int u  = gw % 5;
    const int g0 = 2 * u;
    const int bstart  = gw / 5;
    const int bstride = wavesTotal / 5;

    // --- Hoisted per-wave setup: this wave always builds the same two 8x8
    // blocks, so preload the 40 basis coefficients it needs into registers.
    // Diagonal element e (0..3) of this lane: q = e*32 + lane,
    //   h = e>>1 (which matrix), p = (e&1)*32 + lane (0..63), i = p>>3, j = p&7
    float bk[4][SUBSP];
    int   sAddr[4];                   // LDS address of diag element (row*16+col)
    int   zAddr[4];                   // LDS address of an off-diag element
    #pragma unroll
    for (int e = 0; e < 4; ++e) {
        const int h = e >> 1;
        const int p = ((e & 1) << 5) + lane;   // coalesced across lanes
        const int i = p >> 3, j = p & 7;
        sAddr[e] = (h * 8 + i) * 16 + (h * 8 + j);
        zAddr[e] = (h * 8 + i) * 16 + ((1 - h) * 8 + j);
        #pragma unroll
        for (int k = 0; k < SUBSP; ++k)
            bk[e][k] = basis[(g0 + h) * (SUBSP * 64) + k * 64 + p];
    }

    for (int b = bstart; b < BATCH; b += bstride) {
        // Prefetch next batch row's z while we work on this one.
        if (b + bstride < BATCH)
            __builtin_prefetch(&z[(size_t)(b + bstride) * ZCOLS], 0, 1);

        // 20 z coefficients for the two matrices of this tile.
        float zlo[SUBSP], zhi[SUBSP];
        #pragma unroll
        for (int k = 0; k < SUBSP; ++k) {
            zlo[k] = z[b * ZCOLS +  g0      * SUBSP + k];
            zhi[k] = z[b * ZCOLS + (g0 + 1) * SUBSP + k];
        }

        // Build block-diagonal scaled A (16x16 row-major) in LDS:
        // A = 2^-s * sum_k z_k B_k; off-diagonal blocks zeroed.
        #pragma unroll
        for (int e = 0; e < 4; ++e) {
            float v = 0.0f;
            #pragma unroll
            for (int k = 0; k < SUBSP; ++k)
                v += ((e >> 1) ? zhi[k] : zlo[k]) * bk[e][k];
            S[sAddr[e]] = v * INV_SCALE;
            S[zAddr[e]] = 0.0f;
        }
        wave_lds_fence();

        // A-operand registers (16x4 f32 layout) for the 4 K-chunks; fixed
        // across the whole Taylor recurrence.
        v2f aA[4];
        #pragma unroll
        for (int j = 0; j < 4; ++j) {
            int k0 = 4 * j + 2 * hlf;
            aA[j].x = S[lx * 16 + k0];
            aA[j].y = S[lx * 16 + k0 + 1];
        }

        // First Horner step analytically: T = I + A/N  (D-layout regs).
        float Tcur[8];
        #pragma unroll
        for (int r = 0; r < 8; ++r)
            Tcur[r] = idv[r] + S[(r + 8 * hlf) * 16 + lx] * (1.0f / TAYLOR_N);
        wave_lds_fence();   // done reading A from LDS; reuse S for T
        #pragma unroll
        for (int r = 0; r < 8; ++r)
            S[(r + 8 * hlf) * 16 + lx] = Tcur[r];
        wave_lds_fence();

        // Horner recurrence: T <- I + (A/k) * T, k = N-1 .. 1
        #pragma unroll
        for (int k = TAYLOR_N - 1; k >= 1; --k) {
            v8f acc = {};
            #pragma unroll
            for (int j = 0; j < 4; ++j) {
                v2f bT;
                bT.x = S[(4 * j +     hlf) * 16 + lx];
                bT.y = S[(4 * j + 2 + hlf) * 16 + lx];
                acc = __builtin_amdgcn_wmma_f32_16x16x4_f32(
                        false, aA[j], false, bT, (short)0, acc, false, false);
            }
            const float rk = 1.0f / (float)k;
            wave_lds_fence();
            #pragma unroll
            for (int r = 0; r < 8; ++r)
                S[(r + 8 * hlf) * 16 + lx] = idv[r] + acc[r] * rk;
            wave_lds_fence();
        }

        // Squaring phase: P <- P*P, s times (P lives in LDS tile S).
        #pragma unroll
        for (int sq = 0; sq < SQUARINGS; ++sq) {
            v8f acc = {};
            #pragma unroll
            for (int j = 0; j < 4; ++j) {
                v2f aP, bP;
                int k0 = 4 * j + 2 * hlf;
                aP.x = S[lx * 16 + k0];
                aP.y = S[lx * 16 + k0 + 1];
                bP.x = S[(4 * j +     hlf) * 16 + lx];
                bP.y = S[(4 * j + 2 + hlf) * 16 + lx];
                acc = __builtin_amdgcn_wmma_f32_16x16x4_f32(
                        false, aP, false, bP, (short)0, acc, false, false);
            }
            wave_lds_fence();
            #pragma unroll
            for (int r = 0; r < 8; ++r)
                S[(r + 8 * hlf) * 16 + lx] = acc[r];
            wave_lds_fence();
        }

        // Emit the two 8x8 diagonal blocks, coalesced: 128 floats per tile.
        // out[b, (g0+h)*8 + i, j]  ->  flat b*640 + u*128 + h*64 + i*8 + j
        float* ob = out + (size_t)b * 640 + (size_t)u * 128;
        #pragma unroll
        for (int e = 0; e < 4; ++e) {
            int q = e * 32 + lane;          // 0..127
            int h = q >> 6, p = q & 63;
            int i = p >> 3, j2 = p & 7;
            ob[q] = S[(h * 8 + i) * 16 + (h * 8 + j2)];
        }
        wave_lds_fence();   // keep tile reads ordered before next overwrite
    }
}

extern "C" void kernel_launch(void* const* d_in, const int* in_sizes, int n_in,
                              void* d_out, int out_size, void* d_ws, size_t ws_size,
                              hipStream_t stream) {
    const float* z     = (const float*)d_in[0];   // (65536, 100) f32
    const float* basis = (const float*)d_in[1];   // (100, 8, 8) f32
    float* out = (float*)d_out;                   // (65536, 80, 8) f32
    (void)in_sizes; (void)n_in; (void)out_size; (void)d_ws; (void)ws_size;

    dim3 grid(NBLOCKS), block(256);               // 20480 waves, %5 == 0
    lie_expm_kernel<<<grid, block, 0, stream>>>(z, basis, out);
}